// GreenSFC_90426241450614
// MI455X (gfx1250) — compile-verified
//
#include <hip/hip_runtime.h>
#include <math.h>

// GreenSFC: the T=4.19M-step scan collapses to a scalar affine recurrence
//   h_{t+1} = a*h_t + b  =>  h_t = h* + (h0 - h*) * a^t
// with a ~ 0.98 (contraction). Outputs are three T-length f32 streams derived
// from h_t with a handful of f32 ops. The kernel is pure store-bandwidth bound:
// 50.3 MB / 23.3 TB/s ~= 2.2 us. So: 4 elements per thread, float4 (b128)
// stores, constants precomputed once (setup kernel) and fetched via uniform
// scalar loads. exp2(f64) only runs where a^t hasn't underflowed (t < ~5500),
// i.e. in a handful of early waves (EXECZ-skipped everywhere else).

#define ALPHA1 0.6
#define ALPHA2 0.05

struct GreenSFCConsts {
    double hstar;   // fixed point b/(1-a)
    double dh0;     // h0 - hstar
    double l2a;     // log2(a)
    float  C0f;     // gov_spend + 100*green_investment_prop
    float  Df;      // 1 - ALPHA1*(1-theta)
    float  thetaf;  // clipped tax rate
    float  emisf;   // 1 - sqrt(clip(gip,0,1))
};

// ---- setup: one thread derives all loop-invariant constants into d_ws ----
__global__ void greensfc_setup(const float* __restrict__ tax_rate,
                               const float* __restrict__ gip,
                               const float* __restrict__ gov,
                               GreenSFCConsts* __restrict__ c) {
    if (threadIdx.x != 0 || blockIdx.x != 0) return;

    const double tax = (double)tax_rate[0];
    const double g   = (double)gip[0];
    const double G   = (double)gov[0];

    const double theta = fmin(fmax(tax, 0.05), 0.55);
    const double green = g * 100.0;
    const double C0    = G + green;
    const double D     = 1.0 - ALPHA1 * (1.0 - theta);
    const double coef  = (1.0 - ALPHA1) * (1.0 - theta) / D;  // maps y -> dh
    const double a     = (1.0 - ALPHA2) + ALPHA2 * coef;      // contraction
    const double b     = C0 * coef;
    const double one_m_a = 1.0 - a;                           // > 0 (theta>=0.05)
    const double hstar   = b / one_m_a;

    c->hstar  = hstar;
    c->dh0    = 100.0 - hstar;   // h0 = 100
    c->l2a    = log2(a);         // < 0
    c->C0f    = (float)C0;
    c->Df     = (float)D;
    c->thetaf = (float)theta;
    const float gcl = fminf(fmaxf((float)g, 0.0f), 1.0f);
    c->emisf  = 1.0f - sqrtf(gcl);
}

// ---- main: 4 timesteps per thread, three b128 stores per thread ----
__global__ __launch_bounds__(256) void greensfc_main(
    const GreenSFCConsts* __restrict__ c,
    float* __restrict__ out, long T) {

    // Uniform address -> scalar (s_load) fetches, broadcast to the wave.
    const double hstar  = c->hstar;
    const double dh0    = c->dh0;
    const double l2a    = c->l2a;
    const float  C0f    = c->C0f;
    const float  Df     = c->Df;
    const float  thetaf = c->thetaf;
    const float  emisf  = c->emisf;

    const long t0 = 4L * (blockIdx.x * (long)blockDim.x + threadIdx.x);
    if (t0 >= T) return;

    float4 yv, ev, wv;
    float* yp = reinterpret_cast<float*>(&yv);
    float* ep = reinterpret_cast<float*>(&ev);
    float* wp = reinterpret_cast<float*>(&wv);

#pragma unroll
    for (int j = 0; j < 4; ++j) {
        const double e = (double)(t0 + j) * l2a;   // one f64 mul per element
        double pw = 0.0;
        if (e > -160.0) {                          // EXECZ-skipped for t >~ 5500
            pw = exp2(e);                          // f64 exp2, tiny footprint
        }
        // h_t in closed form; then mirror the scan's per-step f32 arithmetic.
        const float h    = (float)(hstar + dh0 * pw);
        const float y    = (C0f + 0.05f * h) / Df;       // gdp_hist[t]
        const float yd   = y * (1.0f - thetaf);
        const float cons = 0.6f * yd + 0.05f * h;
        const float hnew = h + (yd - cons);              // wealth_hist[t]
        yp[j] = y;
        ep[j] = y * emisf;                               // co2_hist[t]
        wp[j] = hnew;
    }

    // Three 16B stores -> global_store_b128; streams are contiguous per output.
    *reinterpret_cast<float4*>(out + t0)          = yv;
    *reinterpret_cast<float4*>(out + T + t0)      = ev;
    *reinterpret_cast<float4*>(out + 2L * T + t0) = wv;
}

extern "C" void kernel_launch(void* const* d_in, const int* in_sizes, int n_in,
                              void* d_out, int out_size, void* d_ws, size_t ws_size,
                              hipStream_t stream) {
    // Input order (setup_inputs): 0 temp_trajectory, 1 tp_threshold,
    // 2 tp_sensitivity, 3 A_base, 4 gamma, 5 tax_rate,
    // 6 green_investment_prop, 7 gov_spend.
    const float* tax = (const float*)d_in[5];
    const float* gip = (const float*)d_in[6];
    const float* gov = (const float*)d_in[7];
    float* out = (float*)d_out;

    const long T = (long)in_sizes[0];          // 4,194,304
    GreenSFCConsts* consts = (GreenSFCConsts*)d_ws;

    greensfc_setup<<<1, 32, 0, stream>>>(tax, gip, gov, consts);

    const long nthreads = (T + 3) / 4;         // 4 elements per thread
    const int  block    = 256;                 // 8 wave32 per block
    const long grid     = (nthreads + block - 1) / block;  // 4096 blocks
    greensfc_main<<<dim3((unsigned)grid), dim3(block), 0, stream>>>(consts, out, T);
}